// MCPBRNN_constant_OutLoss_53695681134740
// MI455X (gfx1250) — compile-verified
//
#include <hip/hip_runtime.h>
#include <stdint.h>

// ---------------------------------------------------------------------------
// MCPBRNN constant-gate forward == decay scan over x[:, :, 0] + broadcasts.
// 3-phase deterministic scan; per-wave local scan done as a 16x16 triangular
// matmul via 4 chained V_WMMA_F32_16X16X4_F32 (full fp32 precision).
// Decay powers r^e come from a per-block LDS table (no divergent loops).
// ---------------------------------------------------------------------------

typedef __attribute__((ext_vector_type(2))) float v2f;
typedef __attribute__((ext_vector_type(8))) float v8f;

#define SPIN_LEN  1000
#define TRAIN_LEN 500000
#define OBS_BLOCKS 120

struct GateConst {
  float oo, ol, f, f2, f3, r, r16, r256;
};

__device__ __forceinline__ GateConst gates_from(const float* wom, const float* wlm,
                                                const float* wfm) {
  float eo = expf(wom[0]);
  float el = expf(wlm[0]);
  float ef = expf(wfm[0]);
  float inv = 1.0f / (eo + el + ef);
  GateConst g;
  g.oo = eo * inv;
  g.ol = el * inv;
  g.f  = 1.0f - g.oo - g.ol;     // matches reference formula exactly
  g.f2 = g.f * g.f;
  g.f3 = g.f2 * g.f;
  g.r  = g.f2 * g.f2;            // per-row decay f^4
  float t = g.r;
#pragma unroll
  for (int i = 0; i < 4; ++i) t *= t;   // r^16
  g.r16 = t;
#pragma unroll
  for (int i = 0; i < 4; ++i) t *= t;   // r^256
  g.r256 = t;
  return g;
}

// Fill rpow[e] = r^e, e = 0..15 (one thread, fully unrolled).
__device__ __forceinline__ void fill_rpow(float* rpow, float r) {
  if (threadIdx.x == 0) {
    float t = 1.0f;
#pragma unroll
    for (int i = 0; i < 16; ++i) {
      rpow[i] = t;
      t *= r;
    }
  }
}

// Per wave: 256 scan-rows. Loads x (coalesced), builds row values
//   w = f3*u0 + f2*u1 + f*u2 + u3   (end-of-row contribution)
//   p = f2*u0 + f*u1 + u2           (local part of c0)
// into LDS, then computes Y = L @ W with 4 chained WMMA f32 16x16x4:
//   L lower-triangular, L[i][j] = r^(i-j);  W[K][N] = w[16N + K].
// Column N of Y = inclusive decay-scan of segment N (zero carry-in).
__device__ __forceinline__ v8f wave_scan_tile(
    const float* __restrict__ x, long long tl, long long Bv, long long tileBase,
    const GateConst& g, const float* __restrict__ rpow,
    float* __restrict__ ldsW, float* __restrict__ ldsP) {
  const int lane = threadIdx.x & 31;
  const int half = lane >> 4;
  const int m    = lane & 15;

#pragma unroll
  for (int j = 0; j < 8; ++j) {
    int rho = lane + 32 * j;
    long long rs = tileBase + rho;
    float w = 0.0f, p = 0.0f;
    if (rs < Bv) {
      const float4* xv = reinterpret_cast<const float4*>(x) + ((tl + rs) << 1);
      float4 lo = xv[0];   // x[b,0,0], x[b,0,1], x[b,1,0], x[b,1,1]
      float4 hi = xv[1];   // x[b,2,0], x[b,2,1], x[b,3,0], x[b,3,1]
      w = g.f3 * lo.x + g.f2 * lo.z + g.f * hi.x + hi.z;
      p = g.f2 * lo.x + g.f * lo.z + hi.x;
    }
    ldsW[rho] = w;
    if (ldsP) ldsP[rho] = p;
  }
  __builtin_amdgcn_wave_barrier();  // same-wave LDS ops are in-order; fence scheduler

  v8f acc = {0.f, 0.f, 0.f, 0.f, 0.f, 0.f, 0.f, 0.f};
#pragma unroll
  for (int k = 0; k < 4; ++k) {
    v2f A, Bo;
#pragma unroll
    for (int p = 0; p < 2; ++p) {
      // f32 16x4 A layout: lane m (both halves M=0..15), VGPR p holds K = 4k + 2*half + p
      int K = 4 * k + 2 * half + p;
      int d = m - K;
      float av = rpow[d < 0 ? 0 : d];        // table lookup, no loop
      A[p]  = d < 0 ? 0.0f : av;             // L[m][K] = r^(m-K), lower-triangular
      Bo[p] = ldsW[16 * m + K];              // W[K][N=m]
    }
    acc = __builtin_amdgcn_wmma_f32_16x16x4_f32(
        /*neg_a=*/false, A, /*neg_b=*/false, Bo,
        /*c_mod=*/(short)0, acc, /*reuse_a=*/false, /*reuse_b=*/false);
  }
  return acc;  // lane holds Y[M][N], N = lane&15, M = 8*(lane>>4)+v
}

// Phase 1: per-block (2048 rows = 8 wave-tiles) scan aggregate with zero carry.
__global__ void __launch_bounds__(256) k_block_agg(
    const float* __restrict__ x, const float* wom, const float* wlm, const float* wfm,
    const int* tlp, float* __restrict__ ws_agg, long long B) {
  GateConst g = gates_from(wom, wlm, wfm);
  long long tl = (long long)tlp[0];
  long long Bv = B - tl;

  __shared__ float rpow[16];
  __shared__ float ldsW[8][256];
  __shared__ float ldsE[8][16];
  __shared__ float ldsAgg[8];

  fill_rpow(rpow, g.r);
  __syncthreads();

  const int wv = threadIdx.x >> 5;
  const int lane = threadIdx.x & 31;
  long long tileBase = (long long)blockIdx.x * 2048 + (long long)wv * 256;

  v8f Y = wave_scan_tile(x, tl, Bv, tileBase, g, rpow, ldsW[wv], nullptr);

  if (lane >= 16) ldsE[wv][lane - 16] = Y[7];  // M=15 segment-end values
  __builtin_amdgcn_wave_barrier();
  if (lane == 0) {
    float a = 0.0f;
#pragma unroll
    for (int N = 0; N < 16; ++N) a = g.r16 * a + ldsE[wv][N];
    ldsAgg[wv] = a;
  }
  __syncthreads();
  if (threadIdx.x == 0) {
    float a = 0.0f;
#pragma unroll
    for (int j = 0; j < 8; ++j) a = g.r256 * a + ldsAgg[j];
    ws_agg[blockIdx.x] = a;
  }
}

// Obs-std partial reduction (deterministic fixed tree, double accumulation).
__global__ void __launch_bounds__(256) k_obs_partial(
    const float* __restrict__ y, long long B, double* __restrict__ part) {
  __shared__ double sh_s[256];
  __shared__ double sh_q[256];
  long long lo = SPIN_LEN;
  long long hi = (long long)TRAIN_LEN < B ? (long long)TRAIN_LEN : B;
  double s = 0.0, q = 0.0;
  for (long long i = lo + (long long)blockIdx.x * blockDim.x + threadIdx.x; i < hi;
       i += (long long)gridDim.x * blockDim.x) {
    double v = (double)y[i];
    s += v;
    q += v * v;
  }
  sh_s[threadIdx.x] = s;
  sh_q[threadIdx.x] = q;
  __syncthreads();
  for (int off = 128; off > 0; off >>= 1) {
    if (threadIdx.x < off) {
      sh_s[threadIdx.x] += sh_s[threadIdx.x + off];
      sh_q[threadIdx.x] += sh_q[threadIdx.x + off];
    }
    __syncthreads();
  }
  if (threadIdx.x == 0) {
    part[2 * blockIdx.x]     = sh_s[0];
    part[2 * blockIdx.x + 1] = sh_q[0];
  }
}

// Phase 2: serial exclusive scan over block aggregates + finalize std.
__global__ void k_carry_and_std(
    const float* wom, const float* wlm, const float* wfm,
    const double* __restrict__ part, int nPart, long long B,
    const float* __restrict__ ws_agg, float* __restrict__ ws_carry, int nChunks,
    float* __restrict__ obs_out) {
  if (blockIdx.x != 0 || threadIdx.x != 0) return;
  double s = 0.0, q = 0.0;
  for (int i = 0; i < nPart; ++i) {
    s += part[2 * i];
    q += part[2 * i + 1];
  }
  long long hi = (long long)TRAIN_LEN < B ? (long long)TRAIN_LEN : B;
  long long n = hi - (long long)SPIN_LEN;
  double sd = 0.0;
  if (n > 1) {
    double mean = s / (double)n;
    double var = (q - (double)n * mean * mean) / (double)(n - 1);
    sd = var > 0.0 ? sqrt(var) : 0.0;
  }
  obs_out[0] = (float)sd;

  GateConst g = gates_from(wom, wlm, wfm);
  double t = (double)g.r;
  for (int i = 0; i < 11; ++i) t *= t;  // r^2048 (underflows to 0 when it should)
  float rB = (float)t;
  float c = 0.0f;
  for (int i = 0; i < nChunks; ++i) {
    ws_carry[i] = c;
    c = rB * c + ws_agg[i];
  }
}

// Phase 3: recompute local scans with carries, emit all 10 outputs.
__global__ void __launch_bounds__(256) k_final(
    const float* __restrict__ x, const float* wom, const float* wlm, const float* wfm,
    const int* tlp, const float* __restrict__ ws_carry,
    const float* __restrict__ obs_in, float* __restrict__ out, long long B) {
  GateConst g = gates_from(wom, wlm, wfm);
  long long tl = (long long)tlp[0];
  long long Bv = B - tl;

  __shared__ float rpow[16];
  __shared__ float ldsW[8][256];
  __shared__ float ldsP[8][256];
  __shared__ float ldsY[8][256];
  __shared__ float ldsE[8][16];
  __shared__ float ldsC[8][16];   // per-wave segment carry table
  __shared__ float ldsAgg[8];

  fill_rpow(rpow, g.r);
  __syncthreads();

  const int wv = threadIdx.x >> 5;
  const int lane = threadIdx.x & 31;
  const int half = lane >> 4;
  const int nseg = lane & 15;
  long long tileBase = (long long)blockIdx.x * 2048 + (long long)wv * 256;

  v8f Y = wave_scan_tile(x, tl, Bv, tileBase, g, rpow, ldsW[wv], ldsP[wv]);

#pragma unroll
  for (int v = 0; v < 8; ++v) {
    int M = 8 * half + v;
    ldsY[wv][16 * nseg + M] = Y[v];  // tile-row 16*seg + M
  }
  if (half == 1) ldsE[wv][nseg] = Y[7];  // segment-end (M=15)
  __builtin_amdgcn_wave_barrier();
  if (lane == 0) {
    float a = 0.0f;
#pragma unroll
    for (int N = 0; N < 16; ++N) a = g.r16 * a + ldsE[wv][N];
    ldsAgg[wv] = a;
  }
  __syncthreads();

  // Lane 0 of each wave builds the exclusive per-segment carry table once.
  if (lane == 0) {
    float c = ws_carry[blockIdx.x];          // carry into this block
    for (int j = 0; j < wv; ++j) c = g.r256 * c + ldsAgg[j];  // carry into this wave
#pragma unroll
    for (int t = 0; t < 16; ++t) {
      ldsC[wv][t] = c;                       // carry into segment t
      c = g.r16 * c + ldsE[wv][t];
    }
  }
  __builtin_amdgcn_wave_barrier();

  const float obsstd = obs_in[0];
  const float oo = g.oo, ol = g.ol, ff = g.f;

#pragma unroll
  for (int j = 0; j < 8; ++j) {
    int rho = lane + 32 * j;                 // coalesced output rows
    long long rs = tileBase + rho;
    if (rs >= Bv) continue;
    int seg = rho >> 4;
    int M = rho & 15;
    float c  = ldsC[wv][seg];                // carry into this segment (table lookup)
    float pw = rpow[M];                      // r^M (table lookup)
    float Sx = (M == 0 ? 0.0f : ldsY[wv][rho - 1]) + pw * c;  // S_excl
    float c0 = g.f3 * Sx + ldsP[wv][rho];
    long long b = tl + rs;
    float h = oo * c0;
    out[b]               = h;         // h_n
    out[B + b]           = c0;        // c_n
    out[2 * B + b]       = ol * c0;   // l_n
    out[3 * B + b]       = 0.0f;      // bp_n
    out[4 * B + b]       = 0.0f;      // Gate_ib
    out[5 * B + b]       = oo;        // Gate_oo
    out[6 * B + b]       = ol;        // Gate_ol
    out[7 * B + b]       = ff;        // Gate_f
    out[8 * B + 2 * b]     = h;       // h_nout[:,0]
    out[8 * B + 2 * b + 1] = obsstd;  // h_nout[:,1]
    out[10 * B + b]      = obsstd;    // obs_std
  }

  // rows < time_lag stay zero in every output
  if (blockIdx.x == 0) {
    for (long long b = threadIdx.x; b < tl; b += blockDim.x) {
      out[b] = 0.0f; out[B + b] = 0.0f; out[2 * B + b] = 0.0f;
      out[3 * B + b] = 0.0f; out[4 * B + b] = 0.0f; out[5 * B + b] = 0.0f;
      out[6 * B + b] = 0.0f; out[7 * B + b] = 0.0f;
      out[8 * B + 2 * b] = 0.0f; out[8 * B + 2 * b + 1] = 0.0f;
      out[10 * B + b] = 0.0f;
    }
  }
}

extern "C" void kernel_launch(void* const* d_in, const int* in_sizes, int n_in,
                              void* d_out, int out_size, void* d_ws, size_t ws_size,
                              hipStream_t stream) {
  const float* x   = (const float*)d_in[0];
  const float* y   = (const float*)d_in[1];
  const float* wom = (const float*)d_in[2];
  const float* wlm = (const float*)d_in[3];
  const float* wfm = (const float*)d_in[4];
  const int*   tl  = (const int*)d_in[6];   // d_in[5] = epoch (unused)

  long long B = (long long)in_sizes[0] / 8;         // x is (B, 4, 2)
  int nChunks = (int)((B + 2047) / 2048);

  // Workspace layout (doubles first for alignment): needs ~6 KB.
  double* obs_part = (double*)d_ws;                       // 2 * OBS_BLOCKS
  float* ws_agg    = (float*)(obs_part + 2 * OBS_BLOCKS); // nChunks
  float* ws_carry  = ws_agg + nChunks;                    // nChunks
  float* obs_std   = ws_carry + nChunks;                  // 1

  k_obs_partial<<<OBS_BLOCKS, 256, 0, stream>>>(y, B, obs_part);
  k_block_agg<<<nChunks, 256, 0, stream>>>(x, wom, wlm, wfm, tl, ws_agg, B);
  k_carry_and_std<<<1, 32, 0, stream>>>(wom, wlm, wfm, obs_part, OBS_BLOCKS, B,
                                        ws_agg, ws_carry, nChunks, obs_std);
  k_final<<<nChunks, 256, 0, stream>>>(x, wom, wlm, wfm, tl, ws_carry, obs_std,
                                       (float*)d_out, B);
}